// SGCRes_36850819400503
// MI455X (gfx1250) — compile-verified
//
#include <hip/hip_runtime.h>

typedef __attribute__((ext_vector_type(2))) float v2f;
typedef __attribute__((ext_vector_type(8))) float v8f;

#define D_FEAT 64
#define C_OUT 64
#define LN_EPS 1e-5f
#define LDS_STRIDE 68   // 64 + 4 pad: B-fragment LDS reads land on distinct bank pairs

// ---------------------------------------------------------------------------
// Stage 0: zero the SpMM accumulator (d_ws is poisoned with 0xAA by harness)
// ---------------------------------------------------------------------------
__global__ void zero_f32(float* __restrict__ p, int n) {
  int i = blockIdx.x * blockDim.x + threadIdx.x;
  if (i < n) p[i] = 0.0f;
}

// ---------------------------------------------------------------------------
// Stage 1: SpMM scatter.  One wave (32 lanes) per edge:
//   lane l handles features 2l, 2l+1  ->  coalesced 256B gather of feat[col]
//   and 256B of global_atomic_add_f32 into h[row].  feat (25.6MB) is
//   L2-resident on MI455X (192MB L2), so both sides stay on-chip.
// ---------------------------------------------------------------------------
__global__ void spmm_scatter(const float* __restrict__ feat,
                             const int* __restrict__ row,
                             const int* __restrict__ col,
                             const float* __restrict__ vals,
                             float* __restrict__ h, int E) {
  int tid = blockIdx.x * blockDim.x + threadIdx.x;
  int e    = tid >> 5;
  int lane = tid & 31;
  if (e >= E) return;
  int r = row[e];
  int c = col[e];
  float a = vals[e];
  const float* fsrc = feat + (size_t)c * D_FEAT + lane * 2;
  float*       hdst = h    + (size_t)r * D_FEAT + lane * 2;
  v2f f = *(const v2f*)fsrc;
  unsafeAtomicAdd(hdst + 0, a * f.x);   // -> global_atomic_add_f32
  unsafeAtomicAdd(hdst + 1, a * f.y);
}

// ---------------------------------------------------------------------------
// Stage 2: fused [16x64] x [64x64] GEMM (fp32 WMMA, K=4 steps) + bias + LayerNorm.
// One wave per 16-node tile; 8 waves / 256-thread block.
// A fragment (h rows) per ISA layout: lanes 0-15 = M, vgpr{0,1} = K{0,1};
// lanes 16-31 = K{2,3}.  B = W^T fragments read from padded LDS.
// D-matrix layout puts row m in one 16-lane half -> LayerNorm reduction is
// 4x shfl_xor (masks 8,4,2,1) entirely in registers.
// ---------------------------------------------------------------------------
__global__ __launch_bounds__(256) void gemm_ln_wmma(
    const float* __restrict__ h, const float* __restrict__ W,
    const float* __restrict__ bias, const float* __restrict__ gamma,
    const float* __restrict__ beta, float* __restrict__ out, int N) {
  __shared__ float lds_w[C_OUT * LDS_STRIDE];   // 17.4 KB of 320 KB/WGP

  // cooperative stage of W[64][64] (16 KB) into LDS, padded stride
  for (int i = threadIdx.x; i < C_OUT * D_FEAT; i += 256) {
    int r = i >> 6;
    int c = i & 63;
    lds_w[r * LDS_STRIDE + c] = W[i];
  }
  __syncthreads();   // executed by all threads before any wave-uniform exit

  int waveId = threadIdx.x >> 5;
  int lane   = threadIdx.x & 31;
  int tile   = blockIdx.x * 8 + waveId;
  int m0     = tile * 16;
  if (m0 >= N) return;           // wave-uniform: EXEC stays all-1s for WMMA

  int half = lane >> 4;          // 0: K lo pair / rows M0-7 ; 1: K hi pair / rows M8-15
  int l16  = lane & 15;
  const float* arow_ptr = h + (size_t)(m0 + l16) * D_FEAT + 2 * half;

  v8f acc[4] = {};               // four 16x16 f32 accumulators -> C = 64 cols

  for (int k0 = 0; k0 < D_FEAT; k0 += 4) {
    v2f a = *(const v2f*)(arow_ptr + k0);
#pragma unroll
    for (int cj = 0; cj < 4; ++cj) {
      // B[k][n] = W[n][k] : lane reads W row (cj*16+l16), K pair (k0 + 2*half)
      v2f b = *(const v2f*)&lds_w[(cj * 16 + l16) * LDS_STRIDE + k0 + 2 * half];
      acc[cj] = __builtin_amdgcn_wmma_f32_16x16x4_f32(
          /*neg_a=*/false, a, /*neg_b=*/false, b,
          /*c_mod=*/(short)0, acc[cj], /*reuse_a=*/false, /*reuse_b=*/false);
    }
  }

  // bias (per output column n = l16 + 16*cj) must be added before LN stats
  float gm[4], bt[4];
#pragma unroll
  for (int cj = 0; cj < 4; ++cj) {
    int n = l16 + 16 * cj;
    float bn = bias[n];
    gm[cj] = gamma[n];
    bt[cj] = beta[n];
#pragma unroll
    for (int r = 0; r < 8; ++r) acc[cj][r] += bn;
  }

  // LayerNorm stats: row (m0 + r + 8*half) lives in vgpr r across this half's
  // 16 lanes; shfl_xor with masks <=8 reduces within each 16-lane group.
  float mean[8], inv[8];
#pragma unroll
  for (int r = 0; r < 8; ++r) {
    float x0 = acc[0][r], x1 = acc[1][r], x2 = acc[2][r], x3 = acc[3][r];
    float s = x0 + x1 + x2 + x3;
    float q = x0 * x0 + x1 * x1 + x2 * x2 + x3 * x3;
#pragma unroll
    for (int m = 8; m >= 1; m >>= 1) {
      s += __shfl_xor(s, m, 32);
      q += __shfl_xor(q, m, 32);
    }
    float mu  = s * (1.0f / C_OUT);
    float var = q * (1.0f / C_OUT) - mu * mu;
    mean[r] = mu;
    inv[r]  = rsqrtf(var + LN_EPS);
  }

  // normalize + affine + store (16 consecutive floats per 16-lane group)
#pragma unroll
  for (int r = 0; r < 8; ++r) {
    int grow = m0 + r + 8 * half;
    float* orow = out + (size_t)grow * C_OUT + l16;
#pragma unroll
    for (int cj = 0; cj < 4; ++cj) {
      orow[16 * cj] = (acc[cj][r] - mean[r]) * inv[r] * gm[cj] + bt[cj];
    }
  }
}

// ---------------------------------------------------------------------------
extern "C" void kernel_launch(void* const* d_in, const int* in_sizes, int n_in,
                              void* d_out, int out_size, void* d_ws, size_t ws_size,
                              hipStream_t stream) {
  const float* feat  = (const float*)d_in[0];
  // d_in[1] = feat_ori: dead code in the reference (omitted there too)
  const int*   row   = (const int*)d_in[2];
  const int*   col   = (const int*)d_in[3];
  const float* vals  = (const float*)d_in[4];
  const float* W     = (const float*)d_in[5];
  const float* bias  = (const float*)d_in[6];
  const float* gamma = (const float*)d_in[7];
  const float* beta  = (const float*)d_in[8];
  float* out = (float*)d_out;
  float* h   = (float*)d_ws;               // N*64 f32 accumulator (25.6 MB)

  int N = in_sizes[0] / D_FEAT;
  int E = in_sizes[2];

  int nh = N * D_FEAT;
  zero_f32<<<(nh + 255) / 256, 256, 0, stream>>>(h, nh);

  long long sp_threads = (long long)E * 32;          // one wave per edge
  int sp_blocks = (int)((sp_threads + 255) / 256);
  spmm_scatter<<<sp_blocks, 256, 0, stream>>>(feat, row, col, vals, h, E);

  int tiles   = (N + 15) / 16;                       // N=100000 -> 6250 full tiles
  int gblocks = (tiles + 7) / 8;
  gemm_ln_wmma<<<gblocks, 256, 0, stream>>>(h, W, bias, gamma, beta, out, N);
}